// EuclideanAngleLossWithOHEM_57818849738798
// MI455X (gfx1250) — compile-verified
//
#include <hip/hip_runtime.h>
#include <stdint.h>

#define NSEG 16
#define NB 8
#define HW (1u << 20)
#define TPB 256
#define PXB 2048u                      // pixels per block in full-image passes
#define FULL_BLOCKS ((NB * HW) / PXB)  // 4096
#define CROSS_COLS 512u
#define CROSS_BLOCKS (HW / CROSS_COLS) // 2048
#define LSTRIDE (CROSS_COLS + 1)       // 513: odd stride -> conflict-free LDS
#define SENTINEL 0xFFFFFFFFu

typedef float v2f __attribute__((ext_vector_type(2)));
typedef float v8f __attribute__((ext_vector_type(8)));

__device__ __forceinline__ float cart2polar_f(float x, float y) {
  const float PI = 3.14159265358979323846f;
  float th = atanf(y / (x + 1e-12f));
  th += (x < 0.0f) ? PI : 0.0f;
  th += ((x > 0.0f) && (y < 0.0f)) ? (2.0f * PI) : 0.0f;
  return th / (2.0f * PI);
}

__device__ __forceinline__ float blockSum256(float v, float* sb) {
  int t = threadIdx.x;
  sb[t] = v;
  __syncthreads();
#pragma unroll
  for (int s = 128; s > 0; s >>= 1) {
    if (t < s) sb[t] += sb[t + s];
    __syncthreads();
  }
  float r = sb[0];
  __syncthreads();
  return r;
}

// ---------------- init: zero counts + radix histograms ----------------
__global__ void k_init(unsigned* z, unsigned n) {
  unsigned i = blockIdx.x * TPB + threadIdx.x;
  if (i < n) z[i] = 0u;
}

// ---------------- per-batch 16-bin histogram of gt (b128 loads) ----------------
__global__ void k_gt_hist(const int* __restrict__ gt, unsigned* __restrict__ counts) {
  __shared__ unsigned h[NSEG];
  int t = threadIdx.x;
  if (t < NSEG) h[t] = 0u;
  __syncthreads();
  int n = blockIdx.x >> 9;  // 512 blocks per batch
  unsigned hwBase = (blockIdx.x & 511) * PXB;
  const int* gtN = gt + (size_t)n * HW;
#pragma unroll
  for (int i = 0; i < 2; i++) {
    unsigned hw = hwBase + (i * TPB + t) * 4u;
    int4 g4 = *(const int4*)(gtN + hw);
    atomicAdd(&h[g4.x & 15], 1u);
    atomicAdd(&h[g4.y & 15], 1u);
    atomicAdd(&h[g4.z & 15], 1u);
    atomicAdd(&h[g4.w & 15], 1u);
  }
  __syncthreads();
  if (t < NSEG && h[t]) atomicAdd(&counts[n * NSEG + t], h[t]);
}

// ---------------- per-batch pix[] weights, k, negAll flag ----------------
__global__ void k_pix(const unsigned* __restrict__ counts, float* __restrict__ pix,
                      unsigned* __restrict__ negAll, unsigned* __restrict__ selK) {
  int n = threadIdx.x;
  if (n >= NB) return;
  unsigned c[NSEG];
  unsigned pos = 0, nseg = 0;
#pragma unroll
  for (int s = 0; s < NSEG; s++) c[s] = counts[n * NSEG + s];
#pragma unroll
  for (int s = 1; s < NSEG; s++) {
    pos += c[s];
    nseg += (c[s] > 0u) ? 1u : 0u;
  }
  float segAve = (float)pos / fmaxf((float)nseg, 1.0f);
#pragma unroll
  for (int s = 0; s < NSEG; s++) pix[n * NSEG + s] = segAve / fmaxf((float)c[s], 1.0f);
  unsigned sumPos = pos;  // weight>0 iff gt>0 (count(gt>0)==pos)
  unsigned sumNeg = HW - sumPos;
  unsigned k3 = 3u * sumPos;
  selK[n] = (k3 < sumNeg) ? k3 : sumNeg;
  negAll[n] = (pos == 0u) ? 1u : 0u;
}

// ---------------- main elementwise pass (b128 loads/stores) ----------------
__global__ void k_main(const float* __restrict__ pred, const float* __restrict__ gdf,
                       const int* __restrict__ gt, const float* __restrict__ pix,
                       float* __restrict__ term, unsigned* __restrict__ hist0,
                       float* __restrict__ partsTW, float* __restrict__ partsW) {
  __shared__ unsigned h[4096];
  __shared__ float spix[NSEG];
  __shared__ float red[TPB];
  int t = threadIdx.x;
  int n = blockIdx.x >> 9;
  unsigned hwBase = (blockIdx.x & 511) * PXB;
  if (t < NSEG) spix[t] = pix[n * NSEG + t];
  for (int i = t; i < 4096; i += TPB) h[i] = 0u;
  __syncthreads();
  const float* pN = pred + (size_t)n * 2 * HW;
  const float* gN = gdf + (size_t)n * 2 * HW;
  const int* gtN = gt + (size_t)n * HW;
  float ltw = 0.0f, lw = 0.0f;
#pragma unroll
  for (int i = 0; i < 2; i++) {
    unsigned hw = hwBase + (i * TPB + t) * 4u;
    float4 px4 = *(const float4*)(pN + hw);
    float4 py4 = *(const float4*)(pN + HW + hw);
    float4 gx4 = *(const float4*)(gN + hw);
    float4 gy4 = *(const float4*)(gN + HW + hw);
    int4 g4 = *(const int4*)(gtN + hw);
    float pxA[4] = {px4.x, px4.y, px4.z, px4.w};
    float pyA[4] = {py4.x, py4.y, py4.z, py4.w};
    float gxA[4] = {gx4.x, gx4.y, gx4.z, gx4.w};
    float gyA[4] = {gy4.x, gy4.y, gy4.z, gy4.w};
    int gA[4] = {g4.x, g4.y, g4.z, g4.w};
    float tvA[4];
#pragma unroll
    for (int e = 0; e < 4; e++) {
      float dx = pxA[e] - gxA[e], dy = pyA[e] - gyA[e];
      float da = cart2polar_f(gxA[e], gyA[e]) - cart2polar_f(pxA[e], pyA[e]);
      float tv = dx * dx + dy * dy + da * da;
      tvA[e] = tv;
      float w = (gA[e] > 0) ? spix[gA[e] & 15] : 0.0f;
      ltw += tv * w;
      lw += w;
      unsigned key = (w == 0.0f) ? __float_as_uint(tv) : 0u;  // nonneg float: monotone bits
      atomicAdd(&h[key >> 20], 1u);
    }
    float4 tv4;
    tv4.x = tvA[0]; tv4.y = tvA[1]; tv4.z = tvA[2]; tv4.w = tvA[3];
    *(float4*)(term + (size_t)n * HW + hw) = tv4;
  }
  __syncthreads();
  for (int i = t; i < 4096; i += TPB)
    if (h[i]) atomicAdd(&hist0[n * 4096 + i], h[i]);
  float bt = blockSum256(ltw, red);
  float bw = blockSum256(lw, red);
  if (t == 0) {
    partsTW[blockIdx.x] = bt;
    partsW[blockIdx.x] = bw;
  }
}

// ---------------- radix-select scan kernels ----------------
__global__ void k_scan0(const unsigned* __restrict__ hist0, const unsigned* __restrict__ selK,
                        unsigned* __restrict__ selB0, unsigned* __restrict__ selK1,
                        unsigned* __restrict__ selT) {
  int n = threadIdx.x;
  if (n >= NB) return;
  unsigned k = selK[n];
  if (k == 0u) {
    selB0[n] = SENTINEL;
    selK1[n] = 0u;
    selT[n] = SENTINEL;  // nothing kept
    return;
  }
  unsigned cum = 0u;
  for (int b = 4095; b >= 0; b--) {
    unsigned c = hist0[n * 4096 + b];
    if (cum + c >= k) {
      selB0[n] = (unsigned)b;
      selK1[n] = k - cum;
      return;
    }
    cum += c;
  }
  selB0[n] = 0u;
  selK1[n] = 1u;
}

__global__ void k_lvl1(const float* __restrict__ term, const int* __restrict__ gt,
                       const unsigned* __restrict__ negAll, const unsigned* __restrict__ selB0,
                       unsigned* __restrict__ hist1) {
  __shared__ unsigned h[4096];
  int t = threadIdx.x;
  int n = blockIdx.x >> 9;
  unsigned hwBase = (blockIdx.x & 511) * PXB;
  unsigned b0 = selB0[n];
  unsigned na = negAll[n];
  for (int i = t; i < 4096; i += TPB) h[i] = 0u;
  __syncthreads();
  if (b0 != SENTINEL) {
    const float* tN = term + (size_t)n * HW;
    const int* gtN = gt + (size_t)n * HW;
#pragma unroll
    for (int i = 0; i < 2; i++) {
      unsigned hw = hwBase + (i * TPB + t) * 4u;
      float4 tv4 = *(const float4*)(tN + hw);
      int4 g4 = *(const int4*)(gtN + hw);
      float tvA[4] = {tv4.x, tv4.y, tv4.z, tv4.w};
      int gA[4] = {g4.x, g4.y, g4.z, g4.w};
#pragma unroll
      for (int e = 0; e < 4; e++) {
        unsigned key = ((gA[e] == 0) || na) ? __float_as_uint(tvA[e]) : 0u;
        if ((key >> 20) == b0) atomicAdd(&h[(key >> 8) & 0xFFFu], 1u);
      }
    }
  }
  __syncthreads();
  for (int i = t; i < 4096; i += TPB)
    if (h[i]) atomicAdd(&hist1[n * 4096 + i], h[i]);
}

__global__ void k_scan1(const unsigned* __restrict__ hist1, const unsigned* __restrict__ selK1,
                        const unsigned* __restrict__ selB0, unsigned* __restrict__ selB1,
                        unsigned* __restrict__ selK2) {
  int n = threadIdx.x;
  if (n >= NB) return;
  if (selB0[n] == SENTINEL) {
    selB1[n] = SENTINEL;
    selK2[n] = 0u;
    return;
  }
  unsigned k = selK1[n], cum = 0u;
  for (int b = 4095; b >= 0; b--) {
    unsigned c = hist1[n * 4096 + b];
    if (cum + c >= k) {
      selB1[n] = (unsigned)b;
      selK2[n] = k - cum;
      return;
    }
    cum += c;
  }
  selB1[n] = 0u;
  selK2[n] = 1u;
}

__global__ void k_lvl2(const float* __restrict__ term, const int* __restrict__ gt,
                       const unsigned* __restrict__ negAll, const unsigned* __restrict__ selB0,
                       const unsigned* __restrict__ selB1, unsigned* __restrict__ hist2) {
  __shared__ unsigned h[256];
  int t = threadIdx.x;
  int n = blockIdx.x >> 9;
  unsigned hwBase = (blockIdx.x & 511) * PXB;
  unsigned b0 = selB0[n];
  unsigned na = negAll[n];
  if (t < 256) h[t] = 0u;
  __syncthreads();
  if (b0 != SENTINEL) {
    unsigned prefix = (b0 << 12) | selB1[n];
    const float* tN = term + (size_t)n * HW;
    const int* gtN = gt + (size_t)n * HW;
#pragma unroll
    for (int i = 0; i < 2; i++) {
      unsigned hw = hwBase + (i * TPB + t) * 4u;
      float4 tv4 = *(const float4*)(tN + hw);
      int4 g4 = *(const int4*)(gtN + hw);
      float tvA[4] = {tv4.x, tv4.y, tv4.z, tv4.w};
      int gA[4] = {g4.x, g4.y, g4.z, g4.w};
#pragma unroll
      for (int e = 0; e < 4; e++) {
        unsigned key = ((gA[e] == 0) || na) ? __float_as_uint(tvA[e]) : 0u;
        if ((key >> 8) == prefix) atomicAdd(&h[key & 0xFFu], 1u);
      }
    }
  }
  __syncthreads();
  if (t < 256 && h[t]) atomicAdd(&hist2[n * 256 + t], h[t]);
}

__global__ void k_scan2(const unsigned* __restrict__ hist2, const unsigned* __restrict__ selK2,
                        const unsigned* __restrict__ selB0, const unsigned* __restrict__ selB1,
                        unsigned* __restrict__ selT) {
  int n = threadIdx.x;
  if (n >= NB) return;
  if (selB0[n] == SENTINEL) return;  // selT already SENTINEL from scan0
  unsigned k = selK2[n], cum = 0u;
  unsigned b2 = 0u;
  for (int b = 255; b >= 0; b--) {
    unsigned c = hist2[n * 256 + b];
    if (cum + c >= k) {
      b2 = (unsigned)b;
      break;
    }
    cum += c;
  }
  selT[n] = (selB0[n] << 20) | (selB1[n] << 8) | b2;  // exact k-th largest key
}

// ---------------- cross term via V_WMMA_F32_16X16X4_F32 ----------------
// sum_hw (sum_n term) * (sum_m mask)  ==  sum of all entries of Term(8xHW) x Mask^T(HWx8)
__global__ void k_cross(const float* __restrict__ term, const int* __restrict__ gt,
                        const unsigned* __restrict__ negAll, const unsigned* __restrict__ selT,
                        float* __restrict__ partsCross, unsigned* __restrict__ partsMask) {
  __shared__ float ldsT[NB * LSTRIDE];
  __shared__ float ldsM[NB * LSTRIDE];
  __shared__ float waveRed[8];
  __shared__ int redI[TPB];
  int t = threadIdx.x;
  unsigned base = blockIdx.x * CROSS_COLS;
  unsigned tk[NB], na[NB];
#pragma unroll
  for (int r = 0; r < NB; r++) {
    tk[r] = selT[r];
    na[r] = negAll[r];
  }
  int mcnt = 0;
  // stage 1: b128-vectorized fill of term + mask tiles (8 rows x 512 cols)
#pragma unroll
  for (int chunk = 0; chunk < 4; chunk++) {
    unsigned flat = chunk * 1024u + t * 4u;  // 8*512 = 4096 elements total
    unsigned r = flat >> 9;
    unsigned j = flat & 511u;
    size_t g = (size_t)r * HW + base + j;
    float4 tv4 = *(const float4*)(term + g);
    int4 g4 = *(const int4*)(gt + g);
    float tvA[4] = {tv4.x, tv4.y, tv4.z, tv4.w};
    int gA[4] = {g4.x, g4.y, g4.z, g4.w};
#pragma unroll
    for (int e = 0; e < 4; e++) {
      bool rn = (gA[e] == 0) || (na[r] != 0u);
      unsigned key = rn ? __float_as_uint(tvA[e]) : 0u;
      bool m = (key != 0u) && (key >= tk[r]);
      ldsT[r * LSTRIDE + j + e] = tvA[e];
      ldsM[r * LSTRIDE + j + e] = m ? 1.0f : 0.0f;
      mcnt += m ? 1 : 0;
    }
  }
  __syncthreads();

  int lane = t & 31, w = t >> 5;
  int Mrow = lane & 15, hi = lane >> 4;
  float rowSel = (Mrow < 8) ? 1.0f : 0.0f;  // zero-pad rows/cols 8..15 branchlessly
  int rbase = (Mrow & 7) * LSTRIDE;
  v8f c = {0.f, 0.f, 0.f, 0.f, 0.f, 0.f, 0.f, 0.f};
  int hwOff = w * (CROSS_COLS / 8);  // 64 cols per wave
  for (int kk = 0; kk < (int)(CROSS_COLS / 8); kk += 4) {
    int col = rbase + hwOff + kk + 2 * hi;  // A 16x4 f32 layout: lanes 0-15 K={0,1}, 16-31 K={2,3}
    v2f a, b;
    a.x = ldsT[col] * rowSel;
    a.y = ldsT[col + 1] * rowSel;
    b.x = ldsM[col] * rowSel;
    b.y = ldsM[col + 1] * rowSel;
    c = __builtin_amdgcn_wmma_f32_16x16x4_f32(false, a, false, b, (short)0, c, false, false);
  }
  float s = 0.0f;
#pragma unroll
  for (int i = 0; i < 8; i++) s += c[i];
#pragma unroll
  for (int off = 16; off >= 1; off >>= 1) s += __shfl_xor(s, off, 32);
  if (lane == 0) waveRed[w] = s;
  redI[t] = mcnt;
  __syncthreads();
#pragma unroll
  for (int st = 128; st > 0; st >>= 1) {
    if (t < st) redI[t] += redI[t + st];
    __syncthreads();
  }
  if (t == 0) {
    float cs = 0.0f;
#pragma unroll
    for (int i2 = 0; i2 < 8; i2++) cs += waveRed[i2];
    partsCross[blockIdx.x] = cs;
    partsMask[blockIdx.x] = (unsigned)redI[0];
  }
}

// ---------------- final deterministic reduction ----------------
__global__ void k_final(const float* __restrict__ partsTW, const float* __restrict__ partsW,
                        const float* __restrict__ partsCross, const unsigned* __restrict__ partsMask,
                        float* __restrict__ out) {
  __shared__ float red[TPB];
  int t = threadIdx.x;
  float tw = 0.0f, wsum = 0.0f, cross = 0.0f;
  unsigned long long mc = 0ull;
  for (int i = t; i < (int)FULL_BLOCKS; i += TPB) {
    tw += partsTW[i];
    wsum += partsW[i];
  }
  for (int i = t; i < (int)CROSS_BLOCKS; i += TPB) {
    cross += partsCross[i];
    mc += partsMask[i];
  }
  float TW = blockSum256(tw, red);
  float WS = blockSum256(wsum, red);
  float CR = blockSum256(cross, red);
  float MC = blockSum256((float)mc, red);
  if (t == 0) {
    float num = 8.0f * TW + CR;
    float den = 8.0f * (WS + MC);
    out[0] = num / 8.0f / 2.0f / den;
  }
}

// ---------------- host launch ----------------
extern "C" void kernel_launch(void* const* d_in, const int* in_sizes, int n_in,
                              void* d_out, int out_size, void* d_ws, size_t ws_size,
                              hipStream_t stream) {
  const float* pred = (const float*)d_in[0];
  const float* gdf = (const float*)d_in[1];
  const int* gt = (const int*)d_in[2];
  float* out = (float*)d_out;

  uint8_t* w8 = (uint8_t*)d_ws;
  size_t off = 0;
  float* term = (float*)(w8 + off);
  off += (size_t)NB * HW * 4;
  unsigned* zbase = (unsigned*)(w8 + off);
  unsigned* counts = zbase;             // NB*16
  unsigned* hist0 = counts + NB * NSEG; // NB*4096
  unsigned* hist1 = hist0 + NB * 4096;  // NB*4096
  unsigned* hist2 = hist1 + NB * 4096;  // NB*256
  unsigned zcount = NB * NSEG + NB * 4096 * 2 + NB * 256;
  off += (size_t)zcount * 4;
  float* pix = (float*)(w8 + off);
  off += NB * NSEG * 4;
  unsigned* negAll = (unsigned*)(w8 + off);
  off += NB * 4;
  unsigned* selK = (unsigned*)(w8 + off);
  off += NB * 4;
  unsigned* selB0 = (unsigned*)(w8 + off);
  off += NB * 4;
  unsigned* selK1 = (unsigned*)(w8 + off);
  off += NB * 4;
  unsigned* selB1 = (unsigned*)(w8 + off);
  off += NB * 4;
  unsigned* selK2 = (unsigned*)(w8 + off);
  off += NB * 4;
  unsigned* selT = (unsigned*)(w8 + off);
  off += NB * 4;
  float* partsTW = (float*)(w8 + off);
  off += (size_t)FULL_BLOCKS * 4;
  float* partsW = (float*)(w8 + off);
  off += (size_t)FULL_BLOCKS * 4;
  float* partsCross = (float*)(w8 + off);
  off += (size_t)CROSS_BLOCKS * 4;
  unsigned* partsMask = (unsigned*)(w8 + off);
  off += (size_t)CROSS_BLOCKS * 4;

  k_init<<<(zcount + TPB - 1) / TPB, TPB, 0, stream>>>(zbase, zcount);
  k_gt_hist<<<FULL_BLOCKS, TPB, 0, stream>>>(gt, counts);
  k_pix<<<1, 32, 0, stream>>>(counts, pix, negAll, selK);
  k_main<<<FULL_BLOCKS, TPB, 0, stream>>>(pred, gdf, gt, pix, term, hist0, partsTW, partsW);
  k_scan0<<<1, 32, 0, stream>>>(hist0, selK, selB0, selK1, selT);
  k_lvl1<<<FULL_BLOCKS, TPB, 0, stream>>>(term, gt, negAll, selB0, hist1);
  k_scan1<<<1, 32, 0, stream>>>(hist1, selK1, selB0, selB1, selK2);
  k_lvl2<<<FULL_BLOCKS, TPB, 0, stream>>>(term, gt, negAll, selB0, selB1, hist2);
  k_scan2<<<1, 32, 0, stream>>>(hist2, selK2, selB0, selB1, selT);
  k_cross<<<CROSS_BLOCKS, TPB, 0, stream>>>(term, gt, negAll, selT, partsCross, partsMask);
  k_final<<<1, TPB, 0, stream>>>(partsTW, partsW, partsCross, partsMask, out);
}